// GINConv_77214922048104
// MI455X (gfx1250) — compile-verified
//
#include <hip/hip_runtime.h>
#include <hip/hip_bf16.h>

// ---------------------------------------------------------------------------
// GINConv on gfx1250 (MI455X):
//   - scatter-mean aggregation (coalesced gather + L2-resident f32 atomics)
//   - fused 2-layer MLP with V_WMMA_F32_16X16X4_F32
//   - weight panels (<=128KB) staged in LDS via the Tensor Data Mover
//     (tensor_load_to_lds + s_wait_tensorcnt), persistent blocks over M
// ---------------------------------------------------------------------------

typedef __attribute__((ext_vector_type(2))) float        v2f;
typedef __attribute__((ext_vector_type(8))) float        v8f;
typedef __attribute__((ext_vector_type(4))) unsigned int v4u;
typedef __attribute__((ext_vector_type(8))) int          v8i;
typedef __attribute__((ext_vector_type(4))) int          v4i;

#define D_IN    128
#define D_HID   256
#define D_OUT   128

// ---------------------------------------------------------------------------
// 1) Edge scatter: summed[dst] += feat[src] (per channel), deg[dst] += 1.
//    32 lanes per edge; one feat row = 512B = one wave32 of float4 -> coalesced.
// ---------------------------------------------------------------------------
__global__ void gin_scatter_kernel(const float* __restrict__ feat,
                                   const int* __restrict__ src,
                                   const int* __restrict__ dst,
                                   float* __restrict__ summed,
                                   float* __restrict__ deg,
                                   int n_edges) {
    int t = blockIdx.x * blockDim.x + threadIdx.x;
    int e = t >> 5;
    int c = t & 31;
    if (e >= n_edges) return;
    int s = src[e];
    int d = dst[e];
    const float4* frow = (const float4*)(feat + (size_t)s * D_IN);
    float4 v = frow[c];
    float* out = summed + (size_t)d * D_IN + c * 4;
    atomicAdd(out + 0, v.x);
    atomicAdd(out + 1, v.y);
    atomicAdd(out + 2, v.z);
    atomicAdd(out + 3, v.w);
    if (c == 0) atomicAdd(deg + d, 1.0f);
}

// ---------------------------------------------------------------------------
// 2) Finalize in place: summed = (1+eps)*feat + summed / max(deg, 1)
// ---------------------------------------------------------------------------
__global__ void gin_finalize_kernel(const float4* __restrict__ feat4,
                                    const float* __restrict__ eps,
                                    const float* __restrict__ deg,
                                    float4* __restrict__ sum4,
                                    int n4) {
    int t = blockIdx.x * blockDim.x + threadIdx.x;
    if (t >= n4) return;
    int row = t >> 5;                      // 32 float4 per 128-float row
    float inv = 1.0f / fmaxf(deg[row], 1.0f);
    float e = 1.0f + eps[0];
    float4 f = feat4[t];
    float4 s = sum4[t];
    s.x = e * f.x + s.x * inv;
    s.y = e * f.y + s.y * inv;
    s.z = e * f.z + s.z * inv;
    s.w = e * f.w + s.w * inv;
    sum4[t] = s;
}

// ---------------------------------------------------------------------------
// 3) Persistent WMMA fp32 GEMM + bias + ReLU with B resident in LDS.
//      D = relu(A[M,K] * B[K,N] + bias[N]),  K*N*4 <= 128KB (dynamic LDS)
//
//    B is staged once per block by the Tensor Data Mover:
//      D# group0: count=1 | lds_addr | global_addr | type=2
//      D# group1: data_size=4B, tensor_dim0=N, tensor_dim1=K,
//                 tile_dim0=N, tile_dim1=K, tensor_dim0_stride=N
//    Then each wave grid-strides over 16x32 output tiles:
//      A frag (16x4 f32): lane L -> row L%16, VGPR j -> K = 2*(L/16)+j
//      B frag            : lane L -> col L%16, VGPR j -> K = 2*(L/16)+j
//      C/D               : VGPR r -> rows {r, r+8} per lane group
// ---------------------------------------------------------------------------
__global__ void gin_wmma_gemm_lds(const float* __restrict__ A,
                                  const float* __restrict__ B,
                                  const float* __restrict__ bias,
                                  float* __restrict__ D,
                                  int M, int K, int N) {
    extern __shared__ float sB[];          // K*N floats
    const int tid = threadIdx.x;

#if __has_builtin(__builtin_amdgcn_tensor_load_to_lds) && \
    __has_builtin(__builtin_amdgcn_s_wait_tensorcnt)
    if (tid < 32) {                        // wave 0 issues one TDM op
        unsigned long long ga = (unsigned long long)(uintptr_t)B;
        unsigned int lds_off  = (unsigned int)(uintptr_t)(&sB[0]); // LDS byte offset

        v4u g0;
        g0[0] = 1u;                                             // count=1, user mode
        g0[1] = lds_off;                                        // lds_addr [63:32]
        g0[2] = (unsigned int)(ga & 0xFFFFFFFFu);               // global_addr[31:0]
        g0[3] = (unsigned int)((ga >> 32) & 0x01FFFFFFu)        // global_addr[56:32]
              | (2u << 30);                                     // type = 2 ("image")

        unsigned int uN = (unsigned int)N, uK = (unsigned int)K;
        v8i g1;
        g1[0] = (int)(2u << 16);                                // data_size = 4B
        g1[1] = (int)((uN & 0xFFFFu) << 16);                    // tensor_dim0[15:0]
        g1[2] = (int)(((uN >> 16) & 0xFFFFu)                    // tensor_dim0[31:16]
              | ((uK & 0xFFFFu) << 16));                        // tensor_dim1[15:0]
        g1[3] = (int)(((uK >> 16) & 0xFFFFu)                    // tensor_dim1[31:16]
              | ((uN & 0xFFFFu) << 16));                        // tile_dim0 = N
        g1[4] = (int)(uK & 0xFFFFu);                            // tile_dim1 = K rows
        g1[5] = (int)uN;                                        // dim0_stride[31:0] = N
        g1[6] = 0;                                              // stride0 hi, stride1 lo
        g1[7] = 0;                                              // stride1 hi

        v4i gz4 = {0, 0, 0, 0};                                 // groups 2/3 unused (2D)
        v8i gz8 = {0, 0, 0, 0, 0, 0, 0, 0};                     // extra group (clang-23 6-arg form)
        __builtin_amdgcn_tensor_load_to_lds(g0, g1, gz4, gz4, gz8, 0);
        __builtin_amdgcn_s_wait_tensorcnt(0);
    }
#else
    for (int i = tid; i < (K * N) >> 2; i += blockDim.x)
        ((float4*)sB)[i] = ((const float4*)B)[i];
#endif
    __syncthreads();

    const int lane   = tid & 31;
    const int wave   = tid >> 5;
    const int wpb    = blockDim.x >> 5;
    const int tilesN = N >> 5;                      // 32-wide column tiles
    const int totalTiles = (M >> 4) * tilesN;

    const int g   = lane >> 4;                      // lane group 0/1
    const int l16 = lane & 15;

    for (int tile = blockIdx.x * wpb + wave; tile < totalTiles;
         tile += gridDim.x * wpb) {                 // uniform per wave -> EXEC full
        const int tm    = tile / tilesN;
        const int tn    = tile % tilesN;
        const int arow  = tm * 16 + l16;
        const int bcol0 = tn * 32 + l16;

        const float* Arow = A + (size_t)arow * K;

        v8f acc0 = {};
        v8f acc1 = {};

        for (int k = 0; k < K; k += 4) {
            const int ka = k + 2 * g;               // this lane-group's K pair
            v2f a;
            a.x = Arow[ka];
            a.y = Arow[ka + 1];

            v2f b0, b1;
            b0.x = sB[(ka + 0) * N + bcol0];
            b0.y = sB[(ka + 1) * N + bcol0];
            b1.x = sB[(ka + 0) * N + bcol0 + 16];
            b1.y = sB[(ka + 1) * N + bcol0 + 16];

            acc0 = __builtin_amdgcn_wmma_f32_16x16x4_f32(
                false, a, false, b0, (short)0, acc0, false, false);
            acc1 = __builtin_amdgcn_wmma_f32_16x16x4_f32(
                false, a, false, b1, (short)0, acc1, false, false);
        }

        // Store with fused bias + ReLU.
        const int srow = tm * 16 + 8 * g;
        const int scol = tn * 32 + l16;
        const float bb0 = bias[scol];
        const float bb1 = bias[scol + 16];
#pragma unroll
        for (int r = 0; r < 8; ++r) {
            float v0 = acc0[r] + bb0;
            float v1 = acc1[r] + bb1;
            v0 = v0 > 0.0f ? v0 : 0.0f;
            v1 = v1 > 0.0f ? v1 : 0.0f;
            D[(size_t)(srow + r) * N + scol]      = v0;
            D[(size_t)(srow + r) * N + scol + 16] = v1;
        }
    }
}

// ---------------------------------------------------------------------------
// Launch
// ---------------------------------------------------------------------------
extern "C" void kernel_launch(void* const* d_in, const int* in_sizes, int n_in,
                              void* d_out, int out_size, void* d_ws, size_t ws_size,
                              hipStream_t stream) {
    const float* feat = (const float*)d_in[0];   // [N, 128]
    const int*   src  = (const int*)d_in[1];     // [E]
    const int*   dst  = (const int*)d_in[2];     // [E]
    const float* eps  = (const float*)d_in[3];   // [1]
    const float* W1   = (const float*)d_in[4];   // [128, 256]
    const float* b1   = (const float*)d_in[5];   // [256]
    const float* W2   = (const float*)d_in[6];   // [256, 128]
    const float* b2   = (const float*)d_in[7];   // [128]
    float* out = (float*)d_out;                  // [N, 128]

    const int n_nodes = in_sizes[0] / D_IN;      // 40000
    const int n_edges = in_sizes[1];             // 640000

    // Workspace layout: summed[N*128] | deg[N] | h[N*256]
    float* summed = (float*)d_ws;
    float* deg    = summed + (size_t)n_nodes * D_IN;
    float* h      = deg + n_nodes;

    const size_t zero_bytes = ((size_t)n_nodes * D_IN + n_nodes) * sizeof(float);
    (void)hipMemsetAsync(d_ws, 0, zero_bytes, stream);

    // 1) scatter-add edges
    {
        long long threads = (long long)n_edges * 32;
        int block = 256;
        long long grid = (threads + block - 1) / block;
        gin_scatter_kernel<<<(unsigned)grid, block, 0, stream>>>(
            feat, src, dst, summed, deg, n_edges);
    }

    // 2) finalize rst in place
    {
        int n4 = n_nodes * (D_IN / 4);
        int block = 256;
        int grid = (n4 + block - 1) / block;
        gin_finalize_kernel<<<grid, block, 0, stream>>>(
            (const float4*)feat, eps, deg, (float4*)summed, n4);
    }

    const int block   = 256;                       // 8 waves
    const int nBlocks = 512;                       // persistent blocks

    // 3) h = relu(rst @ W1 + b1)   [40000,128] x [128,256], W1 in LDS (128KB)
    {
        size_t shmem = (size_t)D_IN * D_HID * sizeof(float);
        gin_wmma_gemm_lds<<<nBlocks, block, shmem, stream>>>(
            summed, W1, b1, h, n_nodes, D_IN, D_HID);
    }

    // 4) out = relu(h @ W2 + b2)   [40000,256] x [256,128], W2 in LDS (128KB)
    {
        size_t shmem = (size_t)D_HID * D_OUT * sizeof(float);
        gin_wmma_gemm_lds<<<nBlocks, block, shmem, stream>>>(
            h, W2, b2, out, n_nodes, D_HID, D_OUT);
    }
}